// GraphAttentionLayer_82085414961157
// MI455X (gfx1250) — compile-verified
//
#include <hip/hip_runtime.h>
#include <cstdint>

#define N_NODES 4096
#define F_DIM   32
#define U_DIM   32
#define NHEADS  2
#define TI      16            // rows per block
#define STRIDE  4100          // padded LDS row stride (floats) to dodge bank conflicts
#define NEGC    (-1.0e10f)

typedef float v2f __attribute__((ext_vector_type(2)));
typedef float v8f __attribute__((ext_vector_type(8)));
typedef int   v4i __attribute__((ext_vector_type(4)));
typedef v4i __attribute__((address_space(1))) g_v4i;   // global-AS 4-int vector
typedef v4i __attribute__((address_space(3))) l_v4i;   // LDS-AS 4-int vector

#if defined(__HIP_DEVICE_COMPILE__) && !__has_builtin(__builtin_amdgcn_wmma_f32_16x16x4_f32)
#error "__builtin_amdgcn_wmma_f32_16x16x4_f32 not available on this toolchain"
#endif

static __device__ __forceinline__ v8f wmma4(v2f a, v2f b, v8f c) {
#if defined(__HIP_DEVICE_COMPILE__)
  // D = A(16x4,f32) * B(4x16,f32) + C(16x16,f32)
  return __builtin_amdgcn_wmma_f32_16x16x4_f32(false, a, false, b, (short)0, c, false, false);
#else
  (void)a; (void)b;
  return c;   // host parse only; never executed
#endif
}

static __device__ __forceinline__ float leaky(float x) {
  return x >= 0.0f ? x : 0.2f * x;
}

// ---- workspace layout (in floats) ----
#define XSZ  (NHEADS * N_NODES * U_DIM)        // 262144 per (real|imag)
#define SOFF (2 * XSZ)                          // s = X . a1 : [t][h][n]
#define NOFF (SOFF + 2 * NHEADS * N_NODES)      // n = X . a2 : [t][h][n]

// ============================================================================
// Kernel 1a: X{r,i}[h] = H{r,i} @ W[h]   via f32 WMMA 16x16x4
// grid = 256 (16-row blocks), block = 128 (4 waves: wave -> (h, r|i))
// ============================================================================
__global__ __launch_bounds__(128, 4) void xform_kernel(
    const float* __restrict__ Hr, const float* __restrict__ Hi,
    const float* __restrict__ W, float* __restrict__ ws) {
  const int i0   = blockIdx.x * TI;
  const int tid  = threadIdx.x;
  const int w    = tid >> 5, lane = tid & 31;
  const int h    = w & 1, x = (w >> 1) & 1;
  const int M    = lane & 15, kh = lane >> 4, nc = lane & 15;
  const float* Hsrc = x ? Hi : Hr;
  const float* Wh   = W + h * (F_DIM * U_DIM);

  v8f acc0 = {}, acc1 = {};
  for (int k0 = 0; k0 < F_DIM; k0 += 4) {
    const int kb = k0 + 2 * kh;
    v2f av = { Hsrc[(i0 + M) * F_DIM + kb], Hsrc[(i0 + M) * F_DIM + kb + 1] };
    v2f b0 = { Wh[kb * U_DIM + nc],        Wh[(kb + 1) * U_DIM + nc] };
    v2f b1 = { Wh[kb * U_DIM + 16 + nc],   Wh[(kb + 1) * U_DIM + 16 + nc] };
    acc0 = wmma4(av, b0, acc0);
    acc1 = wmma4(av, b1, acc1);
  }
  float* Xout = ws + x * XSZ + h * (N_NODES * U_DIM);
  for (int e = 0; e < 8; ++e) {
    const int row = i0 + e + 8 * kh;              // C/D layout: M = e + 8*(lane/16)
    Xout[row * U_DIM + nc]      = acc0[e];
    Xout[row * U_DIM + 16 + nc] = acc1[e];
  }
}

// ============================================================================
// Kernel 1b: attention scalars s = X.a1, n = X.a2  (tiny; scalar f32)
// ============================================================================
__global__ __launch_bounds__(128) void score_kernel(
    const float* __restrict__ a1, const float* __restrict__ a2,
    float* __restrict__ ws) {
  const int idx = blockIdx.x * blockDim.x + threadIdx.x;   // 16384 tasks
  const int n  = idx & (N_NODES - 1);
  const int hx = idx >> 12;
  const int h  = hx & 1, x = hx >> 1;
  const float* Xrow = ws + x * XSZ + h * (N_NODES * U_DIM) + n * U_DIM;
  float s1 = 0.0f, s2 = 0.0f;
  for (int u = 0; u < U_DIM; ++u) {
    const float xv = Xrow[u];
    s1 += xv * a1[h * U_DIM + u];
    s2 += xv * a2[h * U_DIM + u];
  }
  ws[SOFF + (x * 2 + h) * N_NODES + n] = s1;
  ws[NOFF + (x * 2 + h) * N_NODES + n] = s2;
}

// ============================================================================
// Kernel 2: fused masked-softmax attention + alpha@X with f32 WMMA.
// One WG per 16-row block; A slab (16x4096 f32, 256KB) lives in LDS,
// pulled in with async global->LDS loads (A read from HBM exactly once).
// ============================================================================
#if __has_builtin(__builtin_amdgcn_global_load_async_to_lds_b128)
#define HAVE_ASYNC_LDS 1
#else
#define HAVE_ASYNC_LDS 0
#endif

__global__ __launch_bounds__(512, 1) void attn_kernel(
    const float* __restrict__ A, const float* __restrict__ ws_c,
    float* __restrict__ out) {
  extern __shared__ float slab[];            // TI*STRIDE floats = 262400 B
  __shared__ float smx[64];                  // [h][t][M] row max
  __shared__ float sinv[64];                 // [h][t][M] 1/rowsum
  __shared__ float red8[1024];               // partial-reduction scratch

  const int i0  = blockIdx.x * TI;
  const int tid = threadIdx.x;
  const float* Xr = ws_c;
  const float* Xi = ws_c + XSZ;
  const float* Sp = ws_c + SOFF;
  const float* Np = ws_c + NOFF;

  // ---- stage A slab into LDS ----
  {
    const float* Abase = A + (size_t)i0 * N_NODES;
#if HAVE_ASYNC_LDS
    for (int it = 0; it < 32; ++it) {
      const int idx = tid + it * 512;            // 16384 float4 chunks
      const int row = idx >> 10, c4 = idx & 1023;
      const float* g = Abase + row * N_NODES + c4 * 4;
      float*       l = &slab[row * STRIDE + c4 * 4];
      __builtin_amdgcn_global_load_async_to_lds_b128(
          (g_v4i*)(uintptr_t)g,
          (l_v4i*)(unsigned)(uintptr_t)l,
          0, 0);
    }
#if __has_builtin(__builtin_amdgcn_s_wait_asynccnt)
    __builtin_amdgcn_s_wait_asynccnt(0);
#else
    asm volatile("s_wait_asynccnt 0" ::: "memory");
#endif
#else
    for (int it = 0; it < 32; ++it) {
      const int idx = tid + it * 512;
      const int row = idx >> 10, c4 = idx & 1023;
      *(float4*)&slab[row * STRIDE + c4 * 4] =
          *(const float4*)(Abase + row * N_NODES + c4 * 4);
    }
#endif
  }
  __syncthreads();

  // ---- pass A: per-row softmax max & sum, per (h, t, row). 64 tasks x 8 threads
  {
    const int task = tid >> 3, sub = tid & 7;
    const int Ma = task & 15, ta = (task >> 4) & 1, ha = task >> 5;
    const float sv = Sp[(ta * 2 + ha) * N_NODES + i0 + Ma];
    const float* np   = Np + (ta * 2 + ha) * N_NODES;
    const float* srow = &slab[Ma * STRIDE];

    float m = -3.0e38f;
    for (int k = sub; k < N_NODES; k += 8) {
      const float e = leaky(sv + np[k]) + NEGC * (1.0f - srow[k]);
      m = fmaxf(m, e);
    }
    red8[task * 8 + sub] = m;
    __syncthreads();
    if (sub == 0) {
      float mm = red8[task * 8];
      for (int r = 1; r < 8; ++r) mm = fmaxf(mm, red8[task * 8 + r]);
      smx[task] = mm;
    }
    __syncthreads();
    const float mx = smx[task];
    float s = 0.0f;
    for (int k = sub; k < N_NODES; k += 8) {
      const float e = leaky(sv + np[k]) + NEGC * (1.0f - srow[k]);
      s += __expf(e - mx);
    }
    red8[task * 8 + sub] = s;
    __syncthreads();
    if (sub == 0) {
      float ss = 0.0f;
      for (int r = 0; r < 8; ++r) ss += red8[task * 8 + r];
      sinv[task] = 1.0f / ss;
    }
    __syncthreads();
  }

  // ---- pass B: alpha @ X via f32 WMMA 16x16x4.
  // wave w -> (h = w&1, t = (w>>1)&1, j-quarter p = w>>2); 4 accumulators each.
  const int w    = tid >> 5, lane = tid & 31;
  const int h2   = w & 1, t2 = (w >> 1) & 1, p = w >> 2;
  const int M2   = lane & 15, kh = lane >> 4, nc = lane & 15;
  const int task2 = (h2 << 5) | (t2 << 4) | M2;
  {
    const float sv2  = Sp[(t2 * 2 + h2) * N_NODES + i0 + M2];
    const float* np2 = Np + (t2 * 2 + h2) * N_NODES;
    const float mx2  = smx[task2];
    const float inv2 = sinv[task2];
    const float* Xrh = Xr + h2 * (N_NODES * U_DIM);
    const float* Xih = Xi + h2 * (N_NODES * U_DIM);

    v8f acc0 = {}, acc1 = {}, acc2 = {}, acc3 = {};
    const int jend = p * 1024 + 1024;
    for (int j0 = p * 1024; j0 < jend; j0 += 4) {
      const int jb = j0 + 2 * kh;                     // A & B layouts: K = v + 2*(lane/16)
      const float aA0 = slab[M2 * STRIDE + jb];
      const float aA1 = slab[M2 * STRIDE + jb + 1];
      const float e0 = leaky(sv2 + np2[jb])     + NEGC * (1.0f - aA0);
      const float e1 = leaky(sv2 + np2[jb + 1]) + NEGC * (1.0f - aA1);
      v2f av  = { __expf(e0 - mx2) * inv2, __expf(e1 - mx2) * inv2 };
      v2f br0 = { Xrh[jb * U_DIM + nc],        Xrh[(jb + 1) * U_DIM + nc] };
      v2f br1 = { Xrh[jb * U_DIM + 16 + nc],   Xrh[(jb + 1) * U_DIM + 16 + nc] };
      v2f bi0 = { Xih[jb * U_DIM + nc],        Xih[(jb + 1) * U_DIM + nc] };
      v2f bi1 = { Xih[jb * U_DIM + 16 + nc],   Xih[(jb + 1) * U_DIM + 16 + nc] };
      acc0 = wmma4(av, br0, acc0);
      acc1 = wmma4(av, br1, acc1);
      acc2 = wmma4(av, bi0, acc2);
      acc3 = wmma4(av, bi1, acc3);
    }

    __syncthreads();                 // everyone done reading the A slab
    float* red = slab;               // alias dead slab for partial reduction
    const int base = w * 1024;
    for (int e = 0; e < 8; ++e) {
      const int row = e + 8 * kh;    // C/D layout
      red[base + 0 * 256 + row * 16 + nc] = acc0[e];
      red[base + 1 * 256 + row * 16 + nc] = acc1[e];
      red[base + 2 * 256 + row * 16 + nc] = acc2[e];
      red[base + 3 * 256 + row * 16 + nc] = acc3[e];
    }
  }
  __syncthreads();

  // ---- combine: Hr = a1@Xr - a2@Xi ; Hi = a1@Xi + a2@Xr ; write output
  {
    const float* red = slab;
    for (int idx = tid; idx < TI * NHEADS * U_DIM; idx += 512) {
      const int u  = idx & 31;
      const int hh = (idx >> 5) & 1;
      const int Mr = idx >> 6;
      const int ut = u >> 4, ncc = u & 15;
      const int off = Mr * 16 + ncc;
      float p1r = 0.f, p1i = 0.f, p2r = 0.f, p2i = 0.f;
      for (int pp = 0; pp < 4; ++pp) {
        const int w0 = pp * 4 + 0 + hh;       // t=0 (alpha1)
        const int w1 = pp * 4 + 2 + hh;       // t=1 (alpha2)
        p1r += red[w0 * 1024 + (0 + ut) * 256 + off];
        p1i += red[w0 * 1024 + (2 + ut) * 256 + off];
        p2r += red[w1 * 1024 + (0 + ut) * 256 + off];
        p2i += red[w1 * 1024 + (2 + ut) * 256 + off];
      }
      const int row = i0 + Mr;
      out[row * (NHEADS * U_DIM) + hh * U_DIM + u]                        = p1r - p2i;
      out[N_NODES * NHEADS * U_DIM + row * (NHEADS * U_DIM) + hh * U_DIM + u] = p1i + p2r;
    }
  }
}

// ============================================================================
extern "C" void kernel_launch(void* const* d_in, const int* in_sizes, int n_in,
                              void* d_out, int out_size, void* d_ws, size_t ws_size,
                              hipStream_t stream) {
  (void)in_sizes; (void)n_in; (void)out_size; (void)ws_size;
  const float* Hr = (const float*)d_in[0];
  const float* Hi = (const float*)d_in[1];
  const float* A  = (const float*)d_in[2];
  const float* W  = (const float*)d_in[3];
  const float* a1 = (const float*)d_in[4];
  const float* a2 = (const float*)d_in[5];
  float* out = (float*)d_out;
  float* ws  = (float*)d_ws;

  xform_kernel<<<N_NODES / TI, 128, 0, stream>>>(Hr, Hi, W, ws);
  score_kernel<<<(2 * NHEADS * N_NODES) / 128, 128, 0, stream>>>(a1, a2, ws);
  attn_kernel<<<N_NODES / TI, 512, (size_t)(TI * STRIDE * sizeof(float)), stream>>>(A, ws, out);
}